// MatrixSkipgram_47330539602036
// MI455X (gfx1250) — compile-verified
//
#include <hip/hip_runtime.h>

// 16x16x4 fp32 WMMA fragment types (wave32):
// A: 16x4 f32  -> 64 vals / 32 lanes = 2 VGPRs
// B: 4x16 f32  -> 64 vals / 32 lanes = 2 VGPRs
// C/D: 16x16 f32 -> 256 vals / 32 lanes = 8 VGPRs
typedef float v2f __attribute__((ext_vector_type(2)));
typedef float v8f __attribute__((ext_vector_type(8)));

#define EMBED 100
#define WAVES_PER_BLOCK 8

__global__ __launch_bounds__(32 * WAVES_PER_BLOCK)
void skipgram_wmma_f32(const int* __restrict__ Xarg,
                       const int* __restrict__ Xfun,
                       const int* __restrict__ Xctx,
                       const float* __restrict__ noun,     // [50000, 100]
                       const float* __restrict__ func,     // [10000, 10000]
                       const float* __restrict__ ctxt,     // [50000, 100]
                       float* __restrict__ out, int batch)
{
    const int lane = threadIdx.x & 31;
    const int wave = threadIdx.x >> 5;          // wave-uniform
    const int b = blockIdx.x * WAVES_PER_BLOCK + wave;
    if (b >= batch) return;                     // wave-uniform branch: EXEC stays all-1s

    const int ia = Xarg[b];
    const int ifn = Xfun[b];
    const int ic = Xctx[b];

    const float* __restrict__ M  = func + (size_t)ifn * (EMBED * EMBED);
    const float* __restrict__ av = noun + (size_t)ia * EMBED;
    const float* __restrict__ cv = ctxt + (size_t)ic * EMBED;

    const int g  = lane >> 4;   // half-wave group: selects K pair (A) / K rows (B), M-half (C)
    const int lm = lane & 15;   // A-matrix row within tile / C column

    float acc = 0.0f;

    // 100 rows -> 7 tiles of 16 (pad 100..111 by clamping row addr, zero ctx weight)
    #pragma unroll
    for (int t = 0; t < 7; ++t) {
        int row = t * 16 + lm;
        if (row > EMBED - 1) row = EMBED - 1;   // in-bounds clamp; masked out via ctx=0
        const float* __restrict__ Arow = M + (size_t)row * EMBED + 2 * g;

        v8f C = {0.f, 0.f, 0.f, 0.f, 0.f, 0.f, 0.f, 0.f};

        // K = 100 = 25 steps of 4
        #pragma unroll
        for (int kk = 0; kk < 25; ++kk) {
            // A[m][k], k = 4*kk + 2*g + {0,1}  -> one b64 load per lane
            v2f Af = *(const v2f*)(Arow + 4 * kk);
            // B[k][n] = arg[4*kk + k] broadcast to all 16 columns:
            // lane group g, vgpr r holds B[2*g + r][n] = arg[4*kk + 2*g + r]
            v2f Bf = *(const v2f*)(av + 4 * kk + 2 * g);
            // D = A*B + C   (c_mod=0, no neg/reuse)
            C = __builtin_amdgcn_wmma_f32_16x16x4_f32(false, Af, false, Bf,
                                                      (short)0, C, false, false);
        }

        // C vgpr r, group g holds u[t*16 + r + 8*g] (replicated over the 16 lanes).
        // Fold with ctx weight (0 for padded rows), freeing C for the next tile.
        #pragma unroll
        for (int r = 0; r < 8; ++r) {
            int m = t * 16 + r + 8 * g;
            float cw = (m < EMBED) ? cv[m] : 0.0f;
            acc += cw * C[r];
        }
    }

    // lanes 0-15 hold partial over one half of rows, lanes 16-31 the other half
    acc += __shfl_xor(acc, 16, 32);
    if (lane == 0) out[b] = acc;
}

extern "C" void kernel_launch(void* const* d_in, const int* in_sizes, int n_in,
                              void* d_out, int out_size, void* d_ws, size_t ws_size,
                              hipStream_t stream) {
    const int*   Xarg = (const int*)d_in[0];
    const int*   Xfun = (const int*)d_in[1];
    const int*   Xctx = (const int*)d_in[2];
    const float* noun = (const float*)d_in[3];
    const float* func = (const float*)d_in[4];
    const float* ctxt = (const float*)d_in[5];
    float* out = (float*)d_out;

    const int batch = in_sizes[0];
    const int blocks = (batch + WAVES_PER_BLOCK - 1) / WAVES_PER_BLOCK;
    skipgram_wmma_f32<<<blocks, 32 * WAVES_PER_BLOCK, 0, stream>>>(
        Xarg, Xfun, Xctx, noun, func, ctxt, out, batch);
}